// BoxFilter_37864431682451
// MI455X (gfx1250) — compile-verified
//
#include <hip/hip_runtime.h>
#include <stdint.h>

typedef float v2f __attribute__((ext_vector_type(2)));
typedef float v4f __attribute__((ext_vector_type(4)));
typedef float v8f __attribute__((ext_vector_type(8)));

#define H 1024
#define W 1024
#define RAD 4
#define TILE_C 16                    // output cols per wave
#define TILE_R 64                    // output rows per wave
#define XP_ROWS (TILE_R + 2 * RAD)   // 72 halo rows
#define XP_COLS (TILE_C + 2 * RAD)   // 24 halo cols (= 6 x b128 chunks)
#define XP_CHUNKS (XP_COLS / 4)      // 6
#define XP_STRIDE 28                 // words; 112B row stride -> 16B-aligned chunks
#define T_ROWS XP_ROWS               // 72 intermediate rows
#define T_STRIDE 17
#define WAVES_PER_BLOCK 4

__global__ __launch_bounds__(32 * WAVES_PER_BLOCK)
void BoxFilter_wmma_kernel(const float* __restrict__ x, float* __restrict__ out) {
  __shared__ __align__(16) float lds_xp[WAVES_PER_BLOCK][XP_ROWS * XP_STRIDE];
  __shared__ __align__(16) float lds_t [WAVES_PER_BLOCK][T_ROWS * T_STRIDE];

  const int lane = threadIdx.x & 31;
  const int wave = threadIdx.x >> 5;
  const int m    = lane & 15;   // fragment row (A) / column (B,D)
  const int hi   = lane >> 4;   // half-wave selector per ISA f32 layouts

  const int img = blockIdx.z;                                   // N*C images
  const int x0  = (blockIdx.x * WAVES_PER_BLOCK + wave) * TILE_C;
  const int y0  = blockIdx.y * TILE_R;

  const float* img_ptr = x   + (size_t)img * H * W;
  float*       out_ptr = out + (size_t)img * H * W;

  float* Xp = lds_xp[wave];
  float* T  = lds_t[wave];

  // CDNA5 speculative prefetch of next row-tile's halo (global_prefetch_b8)
  if (lane == 0 && y0 + TILE_R < H) {
    __builtin_prefetch(img_ptr + (size_t)(y0 + TILE_R - RAD) * W + x0, 0, 1);
  }

  // ---- Stage 1: halo tile -> LDS via async tensor-path copies -------------
  // Each 16B chunk is either fully in-bounds (async DMA, no VGPR round-trip)
  // or fully out-of-bounds (ds_store_b128 zero). Disjoint addresses, so the
  // ASYNCcnt writes and DScnt writes cannot race.
  for (int idx = lane; idx < XP_ROWS * XP_CHUNKS; idx += 32) {
    const int r  = idx / XP_CHUNKS;
    const int j  = idx - r * XP_CHUNKS;
    const int gy = y0 - RAD + r;
    const int gx = x0 - RAD + 4 * j;          // 16B-aligned (x0 % 16 == 0)
    float* dst = Xp + r * XP_STRIDE + 4 * j;  // 16B-aligned (stride 112B)
    if (gy >= 0 && gy < H && gx >= 0 && gx <= W - 4) {
      const uint32_t lds_addr = (uint32_t)(uintptr_t)dst;       // low 32b = DS offset
      const uint32_t goff     = (uint32_t)((gy * W + gx) * 4);
      asm volatile("global_load_async_to_lds_b128 %0, %1, %2"
                   :: "v"(lds_addr), "v"(goff), "s"(img_ptr) : "memory");
    } else {
      *(v4f*)dst = (v4f){0.0f, 0.0f, 0.0f, 0.0f};
    }
  }
  asm volatile("s_wait_asynccnt 0" ::: "memory");
  // Wave-private LDS + in-order same-wave DS ops: no barrier required.

  // ---- Banded 0/1 fragments (shared by stage 2 B and stage 3 A) -----------
  // band[kc] element (K = 4*kc + 2*hi + v, idx m): 1 iff m <= K <= m + 8
  v2f band[6];
#pragma unroll
  for (int kc = 0; kc < 6; ++kc) {
    const int k0 = kc * 4 + hi * 2;
    band[kc].x = (k0     >= m && k0     <= m + 2 * RAD) ? 1.0f : 0.0f;
    band[kc].y = (k0 + 1 >= m && k0 + 1 <= m + 2 * RAD) ? 1.0f : 0.0f;
  }

  // ---- Stage 2: horizontal 9-wide box sum via V_WMMA_F32_16X16X4_F32 ------
  // T(72x16) = Xp(72x24) * Bh(24x16), Bh[k][j] = (j <= k <= j+8)
  for (int g = 0; g < 5; ++g) {
    v8f acc = {};
#pragma unroll
    for (int kc = 0; kc < 6; ++kc) {
      const int k0  = kc * 4 + hi * 2;
      const int row = g * 16 + m;
      v2f a;
      if (row < XP_ROWS) {                     // group 4 pads rows 72..79 with 0
        a.x = Xp[row * XP_STRIDE + k0];
        a.y = Xp[row * XP_STRIDE + k0 + 1];
      } else {
        a.x = 0.0f; a.y = 0.0f;
      }
      acc = __builtin_amdgcn_wmma_f32_16x16x4_f32(false, a, false, band[kc],
                                                  (short)0, acc, false, false);
    }
    // D element (M,N) lives in lane N+16*(M>=8), VGPR M&7 -> scatter to T
#pragma unroll
    for (int v = 0; v < 8; ++v) {
      int trow = g * 16 + (hi ? v + 8 : v);
      if (trow < T_ROWS) T[trow * T_STRIDE + m] = acc[v];
    }
  }

  // ---- Stage 3: vertical 9-wide box sum + analytic normalization ----------
  // out(16x16) = Bv(16x24) * T(24x16), Bv[i][k] = (i <= k <= i+8)
  for (int t = 0; t < TILE_R / 16; ++t) {
    v8f acc = {};
#pragma unroll
    for (int kc = 0; kc < 6; ++kc) {
      const int k0 = kc * 4 + hi * 2;
      v2f b;
      b.x = T[(t * 16 + k0)     * T_STRIDE + m];
      b.y = T[(t * 16 + k0 + 1) * T_STRIDE + m];
      acc = __builtin_amdgcn_wmma_f32_16x16x4_f32(false, band[kc], false, b,
                                                  (short)0, acc, false, false);
    }
    const int gx = x0 + m;
    const int cw = min(gx + RAD, W - 1) - max(gx - RAD, 0) + 1;
#pragma unroll
    for (int v = 0; v < 8; ++v) {
      int gy = y0 + t * 16 + (hi ? v + 8 : v);
      int ch = min(gy + RAD, H - 1) - max(gy - RAD, 0) + 1;
      out_ptr[(size_t)gy * W + gx] = acc[v] / (float)(ch * cw);
    }
  }
}

extern "C" void kernel_launch(void* const* d_in, const int* in_sizes, int n_in,
                              void* d_out, int out_size, void* d_ws, size_t ws_size,
                              hipStream_t stream) {
  const float* x = (const float*)d_in[0];
  float* out = (float*)d_out;
  const int nc = in_sizes[0] / (H * W);                  // 16 * 3 = 48 images
  dim3 grid(W / (TILE_C * WAVES_PER_BLOCK),              // 16
            H / TILE_R,                                  // 16
            nc);                                         // 48
  dim3 block(32 * WAVES_PER_BLOCK);                      // 4 waves, wave32
  BoxFilter_wmma_kernel<<<grid, block, 0, stream>>>(x, out);
}